// CentroidDistance_49246095016350
// MI455X (gfx1250) — compile-verified
//
#include <hip/hip_runtime.h>

// CDNA5 / gfx1250 wave32 WMMA implementation of pairwise Poincare distance.
// D = A*B + C via v_wmma_f32_16x16x4_f32 (f32 in, f32 acc — matches reference).

typedef __attribute__((ext_vector_type(2))) float v2f;
typedef __attribute__((ext_vector_type(8))) float v8f;

#define DIMK 128
#define EPSF 1e-7f

// ---------------------------------------------------------------- init: zero column accumulator
__global__ void cd_init_kernel(float* __restrict__ acc, int C) {
    int i = blockIdx.x * blockDim.x + threadIdx.x;
    if (i < C) acc[i] = 0.0f;
}

// ---------------------------------------------------------------- main: WMMA GEMM + acosh epilogue
// Block: 256 threads = 8 waves, covering a 64(M) x 32(Ncols) output tile.
// Each wave owns one 16x16 tile: waves (wm=0..3, wn=0..1).
__global__ __launch_bounds__(256) void cd_main_kernel(
        const float* __restrict__ x,      // [N, 128] row-major
        const float* __restrict__ cen,    // [C, 128] row-major
        float* __restrict__ out_dist,     // [N, C]  (d_out + C)
        float* __restrict__ colacc,       // [C] float accumulator in ws
        int N, int C) {
    __shared__ float scol[32];

    const int tid  = threadIdx.x;
    const int lane = tid & 31;
    const int wave = tid >> 5;
    const int wm   = wave & 3;     // M sub-tile 0..3
    const int wn   = wave >> 2;    // N sub-tile 0..1
    const int kh   = lane >> 4;    // half-wave: 0 or 1
    const int lm   = lane & 15;    // row/col within tile

    if (tid < 32) scol[tid] = 0.0f;
    __syncthreads();

    const int m0 = blockIdx.x * 64 + wm * 16;   // tile base row (N % 16 == 0 -> tile fully valid or fully OOB)
    const int c0 = blockIdx.y * 32 + wn * 16;   // tile base col

    if (m0 < N) {  // wave-uniform: EXEC stays all-ones around WMMA
        // A fragment (16x4 f32): lane holds A[m0+lm][k0+2*kh], [k0+2*kh+1]  -> contiguous float2
        // B fragment (4x16 f32): lane holds Ct[k0+2*kh][c0+lm], [+1] = cen[c0+lm][k0+2*kh..] -> contiguous float2
        const float* ap = x   + (size_t)(m0 + lm) * DIMK + 2 * kh;
        const float* bp = cen + (size_t)(c0 + lm) * DIMK + 2 * kh;

        v8f  acc = {};
        float px2 = 0.0f, pc2 = 0.0f;  // partial squared norms, free from the fragments

        #pragma unroll
        for (int k = 0; k < DIMK; k += 4) {
            v2f a = *(const v2f*)(ap + k);
            v2f b = *(const v2f*)(bp + k);
            acc = __builtin_amdgcn_wmma_f32_16x16x4_f32(
                    false, a, false, b, (short)0, acc, false, false);
            px2 += a.x * a.x + a.y * a.y;
            pc2 += b.x * b.x + b.y * b.y;
        }

        // lane l covers k = {0,1 mod 4} (kh=0) or {2,3 mod 4} (kh=1); pair-sum completes the norm.
        float x2 = px2 + __shfl_xor(px2, 16);   // = |x[m0+lm]|^2 in every lane
        float c2 = pc2 + __shfl_xor(pc2, 16);   // = |c[c0+lm]|^2 in every lane

        // D layout: VGPR v, lanes 0-15 -> M=v, lanes 16-31 -> M=v+8; N = lm.
        float colsum = 0.0f;
        #pragma unroll
        for (int v = 0; v < 8; ++v) {
            int   mloc = kh * 8 + v;
            float x2m  = __shfl(x2, mloc);          // x2 for this output's row (held in lane mloc)
            float dot  = acc[v];
            float sq   = fmaxf(x2m + c2 - 2.0f * dot, 0.0f);
            float den  = fmaxf((1.0f - x2m) * (1.0f - c2), EPSF);
            float z    = fmaxf(1.0f + 2.0f * sq / den, 1.0f + EPSF);
            float dd   = __logf(z + sqrtf(z * z - 1.0f));   // arccosh(z)
            colsum += dd;
            // streamed output, never re-read -> non-temporal store
            __builtin_nontemporal_store(dd, &out_dist[(size_t)(m0 + mloc) * C + (c0 + lm)]);
        }

        // column total over the 16 rows of this tile (lanes l and l^16 share column lm)
        colsum += __shfl_xor(colsum, 16);
        if (lane < 16) atomicAdd(&scol[wn * 16 + lm], colsum);  // ds_add_f32
    }
    __syncthreads();

    if (tid < 32) atomicAdd(&colacc[blockIdx.y * 32 + tid], scol[tid]);  // one global atomic/col/block
}

// ---------------------------------------------------------------- finalize: mean over N
__global__ void cd_final_kernel(const float* __restrict__ colacc,
                                float* __restrict__ out_mean, int N, int C) {
    int c = blockIdx.x * blockDim.x + threadIdx.x;
    if (c < C) out_mean[c] = colacc[c] / (float)N;
}

extern "C" void kernel_launch(void* const* d_in, const int* in_sizes, int n_in,
                              void* d_out, int out_size, void* d_ws, size_t ws_size,
                              hipStream_t stream) {
    const float* x   = (const float*)d_in[0];   // node_repr [N,128] f32
    const float* cen = (const float*)d_in[1];   // centroids [C,128] f32
    const int N = in_sizes[0] / DIMK;           // 50000
    const int C = in_sizes[1] / DIMK;           // 512

    float* out = (float*)d_out;                 // [C] mean, then [N,C] dists
    float* acc = (float*)d_ws;                  // C floats of scratch

    cd_init_kernel<<<dim3((C + 255) / 256), dim3(256), 0, stream>>>(acc, C);

    dim3 grid((N + 63) / 64, C / 32);           // 782 x 16 blocks
    cd_main_kernel<<<grid, dim3(256), 0, stream>>>(x, cen, out + C, acc, N, C);

    cd_final_kernel<<<dim3((C + 255) / 256), dim3(256), 0, stream>>>(acc, out, N, C);
}